// PdmTransform_52415780880407
// MI455X (gfx1250) — compile-verified
//
#include <hip/hip_runtime.h>

typedef __attribute__((ext_vector_type(2))) float v2f;
typedef __attribute__((ext_vector_type(4))) float v4f;
typedef __attribute__((ext_vector_type(8))) float v8f;
typedef __attribute__((ext_vector_type(4))) unsigned int u32x4;
typedef __attribute__((ext_vector_type(8))) int i32x8;
typedef __attribute__((ext_vector_type(4))) int i32x4;

#define NB    32
#define LIN   16000
#define NPH   48
#define NTAP  15
#define KP    16            // taps padded 15 -> 16 (tap 15 == 0)
#define WHALF 7
#define LOUT  (NPH * LIN)   // 768000
#define XPS   16032         // padded x row stride (16 front pad + 16000 + 16 back)

// ---------------------------------------------------------------------------
// Kernel 1: torchaudio-equivalent polyphase sinc/Hann tap table (48 x 16).
// ---------------------------------------------------------------------------
__global__ void build_taps_kernel(float* __restrict__ ktab) {
  int i = blockIdx.x * blockDim.x + threadIdx.x;
  if (i >= NPH * KP) return;
  int p = i / KP;
  int j = i % KP;
  float v = 0.0f;
  if (j < NTAP) {
    const float PI = 3.14159265358979323846f;
    float t = (-(float)p * (1.0f / 48.0f) + (float)(j - WHALF)) * 0.99f;
    t = fminf(6.0f, fmaxf(-6.0f, t));
    float w = cosf(t * (PI / 12.0f));
    w *= w;
    float tp = t * PI;
    float s = (t == 0.0f) ? 1.0f : (sinf(tp) / tp);
    v = s * w * 0.99f;
  }
  ktab[i] = v;
}

// ---------------------------------------------------------------------------
// Kernel 2: zero-pad x into (32 x 16032) image: xpad[row][16+i] = x[row][i].
// Removes all bounds checks from the hot WMMA kernel and makes the TDM tile
// unconditionally in-bounds.
// ---------------------------------------------------------------------------
__global__ void prepad_kernel(const float* __restrict__ x, float* __restrict__ xpad) {
  int i = blockIdx.x * blockDim.x + threadIdx.x;
  if (i >= NB * XPS) return;
  int row = i / XPS;
  int j = i - row * XPS;
  int src = j - 16;
  xpad[i] = (src >= 0 && src < LIN) ? x[row * LIN + src] : 0.0f;
}

// ---------------------------------------------------------------------------
// Kernel 3: polyphase resample as full-f32 WMMA GEMM with TDM-staged inputs.
// Per block: TDM DMAs the 2-D tile (32 floats x 32 rows, stride 16032) into
// LDS (tensor_load_to_lds + s_wait_tensorcnt), then each wave computes 4 rows
// x one 16-frame tile: 3 phase-tiles x 4 K-chunks of V_WMMA_F32_16X16X4_F32.
// Lane ends up with 8 consecutive phases -> two float4 stores.
// ---------------------------------------------------------------------------
__global__ void __launch_bounds__(256)
resample_wmma_kernel(const float* __restrict__ xpad,
                     const float* __restrict__ ktab,
                     float* __restrict__ u) {
  __shared__ float xs[NB * 32];           // 32 rows x 32 floats = 4 KB

  const int lane = threadIdx.x & 31;
  const int wave = threadIdx.x >> 5;      // 0..7
  const int f0   = blockIdx.x * 16;       // frame tile base
  const int n    = lane & 15;             // N index (frame within tile)
  const int kh   = lane >> 4;             // 0: K=0,1  1: K=2,3 (16x4 A/B layout)

  if (wave == 0) {
    // ---- Tensor Data Mover descriptor (CDNA5 ISA) ----
    // Tile: 32 x 32 f32 from xpad[row][f0+9 .. f0+40], row stride 16032.
    unsigned long long ga =
        (unsigned long long)(const void*)(xpad + f0 + 9);
    unsigned int ldsa = (unsigned int)(unsigned long long)(void*)xs; // LDS offset = addr[31:0]

    u32x4 g0;
    g0[0] = 1u;                                           // count=1 (user mode)
    g0[1] = ldsa;                                         // lds_addr
    g0[2] = (unsigned int)(ga & 0xffffffffull);           // global_addr[31:0]
    g0[3] = (unsigned int)((ga >> 32) & 0x01ffffffull)    // global_addr[56:32]
          | (2u << 30);                                   // type = 2 ("image")

    const unsigned int TD0 = 0x40000000u;   // tensor_dim0 (large; tile in-bounds)
    const unsigned int TD1 = 64u;           // tensor_dim1 >= 32 rows
    i32x8 g1;
    g1[0] = (int)(2u << 16);                              // data_size=2 (4B)
    g1[1] = (int)((TD0 & 0xffffu) << 16);                 // tensor_dim0[15:0]
    g1[2] = (int)((TD0 >> 16) | ((TD1 & 0xffffu) << 16)); // td0 hi | td1 lo
    g1[3] = (int)((TD1 >> 16) | (32u << 16));             // td1 hi | tile_dim0=32
    g1[4] = 32;                                           // tile_dim1=32, tile_dim2=0
    g1[5] = (int)(unsigned int)XPS;                       // tensor_dim0_stride lo
    g1[6] = 0;                                            // stride hi / dim1_stride
    g1[7] = 0;
    i32x4 gz4 = {0, 0, 0, 0};                             // groups 2/3 unused (2-D)
    i32x8 gz8 = {0, 0, 0, 0, 0, 0, 0, 0};                 // 6-arg form extra group

    __builtin_amdgcn_tensor_load_to_lds(g0, g1, gz4, gz4, gz8, 0);
    __builtin_amdgcn_s_wait_tensorcnt(0);                 // TENSORcnt == 0
  }
  __syncthreads();

#pragma unroll
  for (int rloc = 0; rloc < 4; ++rloc) {
    const int row = wave * 4 + rloc;      // 0..31
    const float* __restrict__ xrow = xs + row * 32;

    // B fragments from LDS: B[k][n] = xpad_window[n + jc*4 + k]
    v2f bfrag[4];
#pragma unroll
    for (int jc = 0; jc < 4; ++jc) {
      const int j0 = jc * 4 + 2 * kh;
      bfrag[jc].x = xrow[n + j0];
      bfrag[jc].y = xrow[n + j0 + 1];
    }

    float* __restrict__ ur = u + (size_t)row * LOUT;

#pragma unroll
    for (int pt = 0; pt < 3; ++pt) {      // 48 phases = 3 tiles of 16
      const int p0 = pt * 16;
      v8f acc = {};
#pragma unroll
      for (int jc = 0; jc < 4; ++jc) {    // K = 15 padded to 16, 4 chunks of 4
        v2f afrag;
        const int j0 = jc * 4 + 2 * kh;
        afrag.x = ktab[(p0 + n) * KP + j0];
        afrag.y = ktab[(p0 + n) * KP + j0 + 1];
        acc = __builtin_amdgcn_wmma_f32_16x16x4_f32(
            false, afrag, false, bfrag[jc], (short)0, acc, false, false);
      }
      // lane holds phases p = p0 + 8*kh + r (r=0..7) for frame f0+n
      size_t t0 = (size_t)(f0 + n) * NPH + (size_t)(p0 + 8 * kh);
      v4f lo = { acc[0], acc[1], acc[2], acc[3] };
      v4f hi = { acc[4], acc[5], acc[6], acc[7] };
      *(v4f*)(ur + t0)     = lo;          // 16B aligned (t0 % 4 == 0)
      *(v4f*)(ur + t0 + 4) = hi;
    }
  }
}

// ---------------------------------------------------------------------------
// Kernel 4: first-order delta-sigma PDM.  Non-linear scan -> reference-exact
// sequential float recurrence; one wave32, lane b owns batch row b.
//   e = err - x ; y = (e <= 0) ; err = e + y
// ---------------------------------------------------------------------------
__global__ void pdm_kernel(const float* __restrict__ u, float* __restrict__ y) {
  const int b = threadIdx.x;              // 32 lanes = 32 batch rows
  const float* __restrict__ ur = u + (size_t)b * LOUT;
  float* __restrict__ yr = y + (size_t)b * LOUT;
  float err = 0.0f;
  for (int nn = 0; nn < LOUT; nn += 8) {
    __builtin_prefetch(ur + nn + 512, 0, 1);   // global_prefetch_b8
    v4f x0 = *(const v4f*)(ur + nn);
    v4f x1 = *(const v4f*)(ur + nn + 4);
    v4f y0, y1;
#pragma unroll
    for (int k = 0; k < 4; ++k) {
      float e = err - x0[k];
      float o = (e <= 0.0f) ? 1.0f : 0.0f;    // y = (x >= err)
      err = e + o;                             // err = y - x + err
      y0[k] = o;
    }
#pragma unroll
    for (int k = 0; k < 4; ++k) {
      float e = err - x1[k];
      float o = (e <= 0.0f) ? 1.0f : 0.0f;
      err = e + o;
      y1[k] = o;
    }
    *(v4f*)(yr + nn)     = y0;
    *(v4f*)(yr + nn + 4) = y1;
  }
}

// ---------------------------------------------------------------------------
// Launch: taps -> prepad -> resample (TDM + WMMA) -> PDM.
// Workspace layout (floats):
//   ws[0 .. 1023]                      : tap table (48 x 16, padded)
//   ws[1024 .. 1024+32*16032)          : padded input image (2.05 MB)
//   ws[514048 .. 514048+32*768000)     : upsampled signal u (98.3 MB)
// ---------------------------------------------------------------------------
extern "C" void kernel_launch(void* const* d_in, const int* in_sizes, int n_in,
                              void* d_out, int out_size, void* d_ws, size_t ws_size,
                              hipStream_t stream) {
  const float* x = (const float*)d_in[0];   // (32, 16000) f32
  float* out = (float*)d_out;               // (32, 768000) f32 in {0,1}
  float* ws  = (float*)d_ws;

  float* ktab = ws;
  float* xpad = ws + 1024;
  float* u    = ws + 1024 + NB * XPS;

  build_taps_kernel<<<3, 256, 0, stream>>>(ktab);
  prepad_kernel<<<(NB * XPS + 255) / 256, 256, 0, stream>>>(x, xpad);
  resample_wmma_kernel<<<LIN / 16, 256, 0, stream>>>(xpad, ktab, u);
  pdm_kernel<<<1, 32, 0, stream>>>(u, out);
}